// receptive_matmul_14903536517432
// MI455X (gfx1250) — compile-verified
//
#include <hip/hip_runtime.h>
#include <hip/hip_bf16.h>

// receptive (banded) attention scores: out[b,h,i,j] = mask(|i-j|<=128) * <Q[b,h,i,:], K[b,h,j,:]>
// B=4 H=12 L=2048 D=64 R=128. Output 805MB f32 -> store-bandwidth bound on MI455X.

#define L_SEQ   2048
#define D_DIM   64
#define BH      48          // B*H
#define RCP     128         // receptive field
#define NTILE   128         // L_SEQ / 16
#define TBAND   17          // band tiles per 16-row block: jb in [ib-8, ib+8]

typedef __attribute__((ext_vector_type(2))) float v2f;
typedef __attribute__((ext_vector_type(4))) float v4f;
typedef __attribute__((ext_vector_type(8))) float v8f;

// ---------------------------------------------------------------------------
// Kernel 1: zero-fill everything OUTSIDE the band tile region with wide
// non-temporal stores (band tiles |ib-jb|<=8 are written by kernel 2).
// ---------------------------------------------------------------------------
__global__ void zero_outside_band_kernel(v4f* __restrict__ out, long n4) {
    long idx    = (long)blockIdx.x * blockDim.x + threadIdx.x;
    long stride = (long)gridDim.x * blockDim.x;
    const v4f z = {0.0f, 0.0f, 0.0f, 0.0f};
    for (; idx < n4; idx += stride) {
        long lin = idx << 2;                 // flat element index (j fastest)
        int  j   = (int)(lin & (L_SEQ - 1)); // float4 stays inside one 16-col tile
        int  i   = (int)((lin >> 11) & (L_SEQ - 1));
        int  d   = (i >> 4) - (j >> 4);      // tile-diagonal distance
        if (d < -8 || d > 8) {
            __builtin_nontemporal_store(z, out + idx);
        }
    }
}

// ---------------------------------------------------------------------------
// Kernel 2: one wave32 per 16x16 band tile. FP32 WMMA accumulate over D=64
// (16 x v_wmma_f32_16x16x4_f32), per-element band mask, NT store full tile.
// ---------------------------------------------------------------------------
__global__ void band_wmma_kernel(const float* __restrict__ Q,
                                 const float* __restrict__ K,
                                 float* __restrict__ out) {
    const int lane   = threadIdx.x & 31;
    const int waveId = threadIdx.x >> 5;                     // 8 waves / block
    const unsigned tileId = blockIdx.x * 8u + waveId;        // uniform per wave

    const unsigned total = (unsigned)BH * NTILE * TBAND;     // 104448
    if (tileId >= total) return;                             // wave-uniform exit

    const int t  = tileId % TBAND;
    const unsigned r0 = tileId / TBAND;
    const int ib = r0 % NTILE;
    const int bh = r0 / NTILE;
    const int jb = ib - 8 + t;
    if (jb < 0 || jb >= NTILE) return;                       // wave-uniform exit

    // EXEC is all-ones for every lane of this wave from here on (WMMA req).
    const float* __restrict__ qbase = Q + ((long)bh * L_SEQ + (long)ib * 16) * D_DIM;
    const float* __restrict__ kbase = K + ((long)bh * L_SEQ + (long)jb * 16) * D_DIM;

    const int mn   = lane & 15;          // A row m / B col n owned by this lane
    const int half = (lane >> 4) & 1;    // K-pair select within the 4-wide K slab

    // A (16x4 f32): lane holds Q[m][4*kk + 2*half + {0,1}]  -> contiguous float2
    // B (4x16 f32): lane holds K[n][4*kk + 2*half + {0,1}]  -> contiguous float2
    const float* __restrict__ qrow = qbase + mn * D_DIM + 2 * half;
    const float* __restrict__ krow = kbase + mn * D_DIM + 2 * half;

    v8f c = {};
#pragma unroll
    for (int kk = 0; kk < D_DIM / 4; ++kk) {
        v2f a = *(const v2f*)(qrow + kk * 4);
        v2f b = *(const v2f*)(krow + kk * 4);
        // (neg_a, A, neg_b, B, c_mod, C, reuse_a, reuse_b)
        c = __builtin_amdgcn_wmma_f32_16x16x4_f32(false, a, false, b,
                                                  (short)0, c, false, false);
    }

    // C/D layout: VGPR r -> row r (lanes 0-15) / row r+8 (lanes 16-31), col = lane&15.
    const int  jcol    = jb * 16 + mn;
    float* __restrict__ obase =
        out + (long)bh * L_SEQ * L_SEQ + jcol;
#pragma unroll
    for (int r = 0; r < 8; ++r) {
        const int irow = ib * 16 + r + half * 8;
        const int diff = irow - jcol;
        const float v  = (diff <= RCP && -diff <= RCP) ? c[r] : 0.0f;
        __builtin_nontemporal_store(v, obase + (long)irow * L_SEQ);
    }
}

extern "C" void kernel_launch(void* const* d_in, const int* in_sizes, int n_in,
                              void* d_out, int out_size, void* d_ws, size_t ws_size,
                              hipStream_t stream) {
    const float* Q = (const float*)d_in[0];   // [4,12,2048,64] f32
    const float* K = (const float*)d_in[1];   // [4,12,2048,64] f32
    float* out     = (float*)d_out;           // [4,12,2048,2048] f32

    // Pass 1: zero the non-band region (tile-aligned), wide NT stores.
    const long n4 = (long)out_size / 4;       // 50,331,648 float4s
    zero_outside_band_kernel<<<8192, 256, 0, stream>>>((v4f*)out, n4);

    // Pass 2: WMMA band tiles (overwrites nothing pass 1 wrote; fully covers
    // all tiles with |ib-jb|<=8 including their masked-out corners).
    const unsigned totalWaves = (unsigned)BH * NTILE * TBAND; // 104448
    const unsigned blocks     = (totalWaves + 7) / 8;         // 8 waves / block
    band_wmma_kernel<<<blocks, 256, 0, stream>>>(Q, K, out);
}